// DistSageConv_21792664060313
// MI455X (gfx1250) — compile-verified
//
#include <hip/hip_runtime.h>

#define NPART 4
#define NSRC  100000
#define NDST  50000
#define NEDGE 640000
#define NOWN  25000
#define DIN   128
#define DOUT  128
#define KDIM  256   // 2*DIN

typedef __attribute__((ext_vector_type(2))) float v2f;
typedef __attribute__((ext_vector_type(8))) float v8f;

// ---------------------------------------------------------------------------
// Phase 0: zero the neighbor-aggregation workspace (ng) with wide stores.
// ---------------------------------------------------------------------------
__global__ void __launch_bounds__(256) zero_ws_kernel(float4* __restrict__ ws,
                                                      long n4) {
  long i = (long)blockIdx.x * blockDim.x + threadIdx.x;
  if (i < n4) ws[i] = make_float4(0.f, 0.f, 0.f, 0.f);
}

// ---------------------------------------------------------------------------
// Phase 1: edge scatter-add.  32 lanes per edge, each lane moves 4 floats
// (float4 coalesced gather -> 4x global_atomic_add_f32 scatter, no return).
// 256-thread block handles 8 edges.
// ---------------------------------------------------------------------------
__global__ void __launch_bounds__(256) scatter_add_kernel(
    const float* __restrict__ x,       // [P, NSRC, DIN]
    const int*   __restrict__ esrc,    // [P, E]
    const int*   __restrict__ edst,    // [P, E]
    float*       __restrict__ ng) {    // [P, NDST, DIN]
  long gid  = (long)blockIdx.x * blockDim.x + threadIdx.x;
  long edge = gid >> 5;                      // global edge over P*E
  int  lane = threadIdx.x & 31;
  if (edge >= (long)NPART * NEDGE) return;
  int p = (int)(edge / NEDGE);
  int e = (int)(edge % NEDGE);
  int s = esrc[(long)p * NEDGE + e];
  int d = edst[(long)p * NEDGE + e];
  const float4* src = (const float4*)(x + ((long)p * NSRC + s) * DIN);
  float4 v = src[lane];                      // 512B per edge row per wave
  float* dst = ng + ((long)p * NDST + d) * DIN + lane * 4;
  atomicAdd(dst + 0, v.x);
  atomicAdd(dst + 1, v.y);
  atomicAdd(dst + 2, v.z);
  atomicAdd(dst + 3, v.w);
}

// ---------------------------------------------------------------------------
// Phase 2: gathered GEMM  out[p] = h[oid] @ W^T + b  via V_WMMA_F32_16X16X4_F32.
//   h row = concat(x[p][self_ids[oid]], ng[p][oid])  (K = 256)
// Block: 256 threads = 8 waves. Each wave computes one 16x16 tile:
//   wave w -> N-tile w (cols w*16..w*16+15), block -> one 16-row M-tile.
// A frag (16x4 f32, 2 VGPR): lane l = row (l&15), K pair 2*(l>>4).. +1
// B frag (4x16  f32, 2 VGPR): lane l = col (l&15), same K pair; B[k][n]=W[n][k]
// C/D  (8 VGPR): row r + 8*(lane>>4), col lane&15
// ---------------------------------------------------------------------------
__global__ void __launch_bounds__(256) sage_gemm_kernel(
    const float* __restrict__ x,         // [P, NSRC, DIN]
    const float* __restrict__ ng,        // [P, NDST, DIN]
    const float* __restrict__ W,         // [DOUT, KDIM]
    const float* __restrict__ bias,      // [DOUT]
    const int*   __restrict__ self_ids,  // [P, NDST]
    const int*   __restrict__ owned_ids, // [P, NOWN]
    float*       __restrict__ out) {     // [P, NOWN, DOUT]
  const int p     = blockIdx.y;
  const int mtile = blockIdx.x;
  const int wave  = threadIdx.x >> 5;
  const int lane  = threadIdx.x & 31;
  const int half  = lane >> 4;           // 0: K+0..1, 1: K+2..3
  const int col   = wave * 16 + (lane & 15);
  const int khalf = half * 2;

  // Per-lane gathered row (clamped; partial tile predicated at store only,
  // so EXEC is all-1s around every WMMA as the ISA requires).
  int row  = mtile * 16 + (lane & 15);
  int rowc = row < NOWN ? row : NOWN - 1;
  int oid  = owned_ids[(long)p * NOWN + rowc];
  int sid  = self_ids[(long)p * NDST + oid];
  const float* hx = x  + ((long)p * NSRC + sid) * DIN;  // h[:,   0:128)
  const float* hn = ng + ((long)p * NDST + oid) * DIN;  // h[:, 128:256)
  const float* wr = W + (long)col * KDIM;               // row of W = col of W^T

  v8f acc = {};
  // K = 0..127 : self features
#pragma unroll 8
  for (int k = 0; k < DIN; k += 4) {
    v2f a = *(const v2f*)(hx + k + khalf);
    v2f b = *(const v2f*)(wr + k + khalf);
    acc = __builtin_amdgcn_wmma_f32_16x16x4_f32(
        /*neg_a=*/false, a, /*neg_b=*/false, b,
        /*c_mod=*/(short)0, acc, /*reuse_a=*/false, /*reuse_b=*/false);
  }
  // K = 128..255 : aggregated neighbor features
#pragma unroll 8
  for (int k = 0; k < DIN; k += 4) {
    v2f a = *(const v2f*)(hn + k + khalf);
    v2f b = *(const v2f*)(wr + DIN + k + khalf);
    acc = __builtin_amdgcn_wmma_f32_16x16x4_f32(
        false, a, false, b, (short)0, acc, false, false);
  }

  // Epilogue: bias + predicated store (C/D layout: VGPR r -> M = r + 8*half).
  float bv = bias[col];
  int rbase = mtile * 16 + half * 8;
#pragma unroll
  for (int r = 0; r < 8; ++r) {
    int orow = rbase + r;
    if (orow < NOWN) {
      out[((long)p * NOWN + orow) * DOUT + col] = acc[r] + bv;
    }
  }
}

// ---------------------------------------------------------------------------
extern "C" void kernel_launch(void* const* d_in, const int* in_sizes, int n_in,
                              void* d_out, int out_size, void* d_ws, size_t ws_size,
                              hipStream_t stream) {
  const float* x    = (const float*)d_in[0];  // [P, NSRC, DIN]
  const float* W    = (const float*)d_in[1];  // [DOUT, 2*DIN]
  const float* b    = (const float*)d_in[2];  // [DOUT]
  const int*   esrc = (const int*)  d_in[3];  // [P, E]
  const int*   edst = (const int*)  d_in[4];  // [P, E]
  const int*   sids = (const int*)  d_in[5];  // [P, NDST]
  const int*   oids = (const int*)  d_in[6];  // [P, NOWN]
  float* out = (float*)d_out;                 // [P, NOWN, DOUT]
  float* ng  = (float*)d_ws;                  // [P, NDST, DIN] scratch

  // Phase 0: zero ng (102.4 MB) with float4 stores.
  long n4 = (long)NPART * NDST * DIN / 4;
  zero_ws_kernel<<<(unsigned)((n4 + 255) / 256), 256, 0, stream>>>(
      (float4*)ng, n4);

  // Phase 1: scatter-add over all P*E edges, 32 lanes/edge.
  long nthreads = (long)NPART * NEDGE * 32;
  scatter_add_kernel<<<(unsigned)((nthreads + 255) / 256), 256, 0, stream>>>(
      x, esrc, edst, ng);

  // Phase 2: WMMA GEMM over owned rows.
  dim3 grid((NOWN + 15) / 16, NPART);
  sage_gemm_kernel<<<grid, 256, 0, stream>>>(x, ng, W, b, sids, oids, out);
}